// UserOnlyGate_12635793784887
// MI455X (gfx1250) — compile-verified
//
#include <hip/hip_runtime.h>
#include <hip/hip_bf16.h>

typedef __attribute__((ext_vector_type(2))) float v2f;
typedef __attribute__((ext_vector_type(8))) float v8f;
typedef __attribute__((ext_vector_type(4))) unsigned int v4u;
typedef __attribute__((ext_vector_type(8))) int v8i;
typedef __attribute__((ext_vector_type(4))) int v4i;

#define GD 256            // user_dim
#define GE 64             // num experts
#define GK 8              // top_k
#define BLOCK_THREADS 256 // 8 waves (wave32)
#define ROWS_PER_BLOCK 128// 8 waves x 16 rows
#define NEG_BIG (-3.0e38f)

// ---- Tensor Data Mover: DMA a [rows x 256] f32 tile from global to LDS ----
// D# per CDNA5 ISA 8.3/8.4: group0 = {count|flags, lds_addr, global_addr_lo,
// global_addr_hi|type=2}; group1 carries data_size / dims / tile / stride.
__device__ __forceinline__ void tdm_load_tile(const float* gsrc, unsigned lds_byte_off, int rows) {
    unsigned long long ga = (unsigned long long)(const void*)gsrc;
    v4u g0;
    g0[0] = 1u;                                          // count=1, user descriptor
    g0[1] = lds_byte_off;                                // lds_addr (bytes)
    g0[2] = (unsigned)(ga & 0xFFFFFFFFu);                // global_addr[31:0]
    g0[3] = (unsigned)((ga >> 32) & 0x01FFFFFFu)         // global_addr[56:32]
          | (2u << 30);                                  // type = 2 ("image")
    v8i g1;
    g1[0] = (2 << 16);                                   // wg_mask=0, data_size=4B
    g1[1] = (int)(256u << 16);                           // tensor_dim0[15:0]=256 @ bits[79:64->31:16]
    g1[2] = (int)(((unsigned)rows & 0xFFFFu) << 16);     // tensor_dim0 hi=0 | tensor_dim1 lo=rows
    g1[3] = (int)(256u << 16);                           // tensor_dim1 hi=0 | tile_dim0=256
    g1[4] = rows;                                        // tile_dim1=rows | tile_dim2=0
    g1[5] = 256;                                         // tensor_dim0_stride lo32 = 256
    g1[6] = 0;                                           // stride hi | dim1_stride lo
    g1[7] = 0;
    v4i z4 = (v4i)0;                                     // groups 2/3 unused (2-D tensor)
    v8i z8 = (v8i)0;                                     // 5th SGPR group (clang-23 form), unused
    __builtin_amdgcn_tensor_load_to_lds(g0, g1, z4, z4, z8, 0);
}

__global__ __launch_bounds__(BLOCK_THREADS)
void moe_gate_topk_kernel(const float* __restrict__ u,
                          const float* __restrict__ W,
                          const float* __restrict__ bvec,
                          float* __restrict__ out,
                          int nrows) {
    extern __shared__ float lds[];
    float* wT   = lds;                       // [GE][GD] W transposed     (64 KB)
    float* buf0 = lds + GE * GD;             // [128][GD] u tile, buffer0 (128 KB)
    float* buf1 = buf0 + ROWS_PER_BLOCK * GD;// [128][GD] u tile, buffer1 (128 KB)
    const unsigned ldsOff0 = (unsigned)(size_t)(void*)buf0;  // LDS byte offsets
    const unsigned ldsOff1 = (unsigned)(size_t)(void*)buf1;

    const int tid  = threadIdx.x;
    const int wave = tid >> 5;
    const int lane = tid & 31;
    const int half = lane >> 4;              // WMMA A/B lane split: K=0,1 vs K=2,3
    const int l16  = lane & 15;

    // ---- Stage W transposed into LDS (once; 64 KB, L2-resident) ----
    for (int i = tid; i < (GD * GE) / 4; i += BLOCK_THREADS) {
        const float4 w4 = reinterpret_cast<const float4*>(W)[i];
        const int d = (4 * i) >> 6;          // k index (row of W)
        const int e = (4 * i) & 63;          // expert  (col of W)
        wT[(e + 0) * GD + d] = w4.x;
        wT[(e + 1) * GD + d] = w4.y;
        wT[(e + 2) * GD + d] = w4.z;
        wT[(e + 3) * GD + d] = w4.w;
    }

    // bias fragment in C layout: expert = eg*16 + l16
    float bb[4];
    #pragma unroll
    for (int eg = 0; eg < 4; ++eg) bb[eg] = bvec[eg * 16 + l16];

    const int numRB = nrows / ROWS_PER_BLOCK;

    // ---- Prologue: TDM first tile into buf0 ----
    if (wave == 0) {
        if ((int)blockIdx.x < numRB)
            tdm_load_tile(u + (size_t)blockIdx.x * ROWS_PER_BLOCK * GD, ldsOff0, ROWS_PER_BLOCK);
        __builtin_amdgcn_s_wait_tensorcnt(0);
    }
    __syncthreads();

    int parity = 0;
    for (int rb = blockIdx.x; rb < numRB; rb += gridDim.x) {
        float* uCur = parity ? buf1 : buf0;
        // ---- Kick off DMA of the next tile into the other buffer ----
        const int rbn = rb + (int)gridDim.x;
        if (wave == 0 && rbn < numRB)
            tdm_load_tile(u + (size_t)rbn * ROWS_PER_BLOCK * GD,
                          parity ? ldsOff0 : ldsOff1, ROWS_PER_BLOCK);

        const size_t rowBase = (size_t)rb * ROWS_PER_BLOCK;

        // ---- GEMM: each wave -> 16 rows x 64 experts via f32 WMMA ----
        // A (16x4): lane(l16,half) holds u[row=l16][4k+2*half .. +1]
        // B (4x16): lane(l16,half) holds W[4k+2*half .. +1][e=eg*16+l16]
        const float* aPtr = uCur + (wave * 16 + l16) * GD + 2 * half;
        const float* bPtr = wT + l16 * GD + 2 * half;

        v8f acc0 = {}, acc1 = {}, acc2 = {}, acc3 = {};
        #pragma unroll 4
        for (int k = 0; k < GD / 4; ++k) {
            const v2f a  = *reinterpret_cast<const v2f*>(aPtr + 4 * k);
            const v2f b0 = *reinterpret_cast<const v2f*>(bPtr + 0 * 16 * GD + 4 * k);
            const v2f b1 = *reinterpret_cast<const v2f*>(bPtr + 1 * 16 * GD + 4 * k);
            const v2f b2 = *reinterpret_cast<const v2f*>(bPtr + 2 * 16 * GD + 4 * k);
            const v2f b3 = *reinterpret_cast<const v2f*>(bPtr + 3 * 16 * GD + 4 * k);
            acc0 = __builtin_amdgcn_wmma_f32_16x16x4_f32(false, a, false, b0, (short)0, acc0, false, false);
            acc1 = __builtin_amdgcn_wmma_f32_16x16x4_f32(false, a, false, b1, (short)0, acc1, false, false);
            acc2 = __builtin_amdgcn_wmma_f32_16x16x4_f32(false, a, false, b2, (short)0, acc2, false, false);
            acc3 = __builtin_amdgcn_wmma_f32_16x16x4_f32(false, a, false, b3, (short)0, acc3, false, false);
        }
        __syncthreads();   // all waves done reading uCur A-fragments

        // ---- Transpose logits to row-major LDS scratch (reuse uCur) ----
        // C layout: acc[eg][r] -> row = wave*16 + r + 8*half, expert = eg*16+l16
        float* scratch = uCur;   // [128][64] floats (32 KB of the 128 KB buffer)
        {
            float* sb = scratch + (wave * 16 + 8 * half) * GE + l16;
            #pragma unroll
            for (int r = 0; r < 8; ++r) {
                sb[r * GE + 0]  = acc0[r] + bb[0];
                sb[r * GE + 16] = acc1[r] + bb[1];
                sb[r * GE + 32] = acc2[r] + bb[2];
                sb[r * GE + 48] = acc3[r] + bb[3];
            }
        }
        __syncthreads();

        // ---- Lane-local epilogue: thread owns half a row (32 experts) ----
        const int myRow  = tid >> 1;         // 0..127
        const int myHalf = (tid & 1) * 32;   // expert offset 0 or 32
        float x[32];
        {
            const float4* sp = reinterpret_cast<const float4*>(scratch + myRow * GE + myHalf);
            #pragma unroll
            for (int j = 0; j < 8; ++j) {
                const float4 q = sp[j];
                x[4 * j + 0] = q.x; x[4 * j + 1] = q.y;
                x[4 * j + 2] = q.z; x[4 * j + 3] = q.w;
            }
        }

        // top-8 threshold over the row's 64 logits: 8 extract-max rounds,
        // only cross-lane op is one xor-1 shuffle per round.
        float t[32];
        #pragma unroll
        for (int j = 0; j < 32; ++j) t[j] = x[j];
        float thr = NEG_BIG, rowMax = NEG_BIG;
        #pragma unroll
        for (int it = 0; it < GK; ++it) {
            float mm = t[0];
            #pragma unroll
            for (int j = 1; j < 32; ++j) mm = fmaxf(mm, t[j]);
            mm = fmaxf(mm, __shfl_xor(mm, 1, 32));      // combine the two half-rows
            thr = mm;
            if (it == 0) rowMax = mm;
            #pragma unroll
            for (int j = 0; j < 32; ++j) t[j] = (t[j] == mm) ? NEG_BIG : t[j];
        }

        // masked exp + renormalize. Softmax denominator cancels:
        // w = exp(v-m)*mask / sum(exp(v-m)*mask); sum >= exp(0) = 1 so EPS is moot.
        float s = 0.0f;
        #pragma unroll
        for (int j = 0; j < 32; ++j) {
            const float w = (x[j] >= thr) ? __expf(x[j] - rowMax) : 0.0f;
            x[j] = w;
            s += w;
        }
        s += __shfl_xor(s, 1, 32);
        const float inv = __builtin_amdgcn_rcpf(s);

        float4* op = reinterpret_cast<float4*>(out + (rowBase + (size_t)myRow) * GE + myHalf);
        #pragma unroll
        for (int j = 0; j < 8; ++j) {
            float4 q;
            q.x = x[4 * j + 0] * inv; q.y = x[4 * j + 1] * inv;
            q.z = x[4 * j + 2] * inv; q.w = x[4 * j + 3] * inv;
            op[j] = q;
        }

        // ---- Wait for the prefetched tile, flip buffers ----
        if (wave == 0) __builtin_amdgcn_s_wait_tensorcnt(0);
        __syncthreads();
        parity ^= 1;
    }
}

extern "C" void kernel_launch(void* const* d_in, const int* in_sizes, int n_in,
                              void* d_out, int out_size, void* d_ws, size_t ws_size,
                              hipStream_t stream) {
    // setup_inputs order: h, u, W, b, top_k
    const float* u  = (const float*)d_in[1];
    const float* W  = (const float*)d_in[2];
    const float* bv = (const float*)d_in[3];
    float* out = (float*)d_out;

    const int N = in_sizes[1] / GD;

    // 64 KB W^T + 2 x 128 KB double-buffered u tiles = 320 KB (full WGP LDS)
    const size_t smem = (size_t)(GE * GD + 2 * ROWS_PER_BLOCK * GD) * sizeof(float);
    (void)hipFuncSetAttribute((const void*)moe_gate_topk_kernel,
                              hipFuncAttributeMaxDynamicSharedMemorySize, (int)smem);

    const int numRB = N / ROWS_PER_BLOCK;   // 4096
    int blocks = numRB < 1024 ? numRB : 1024;

    moe_gate_topk_kernel<<<blocks, BLOCK_THREADS, smem, stream>>>(u, W, bv, out, N);
}